// CreateProposal_7361573945988
// MI455X (gfx1250) — compile-verified
//
#include <hip/hip_runtime.h>
#include <hip/hip_bf16.h>
#include <math.h>
#include <stdint.h>

// ---------------- problem constants ----------------
#define H_FEAT 100
#define W_FEAT 100
#define HWSZ   (H_FEAT * W_FEAT)          // 10000
#define NA     9
#define NPROP  (HWSZ * NA)                // 90000
#define NPAD   131072                     // next pow2 >= NPROP
#define NSEL   6000                       // PRE_NMS_TOP_N
#define NPOST  300                        // POST_NMS_TOP_N
#define NMS_TH 0.7f
#define NW     188                        // ceil(6000/32) words per mask row
#define RB     375                        // 6000/16 row tiles

typedef float v2f __attribute__((ext_vector_type(2)));
typedef float v8f __attribute__((ext_vector_type(8)));
typedef int   v2i __attribute__((ext_vector_type(2)));

typedef __attribute__((address_space(1))) v2i* gptr_v2i;
typedef __attribute__((address_space(3))) v2i* lptr_v2i;

struct AnchorsArg { float v[4 * NA]; };

// ---------------- workspace layout (bytes) ----------------
#define WS_KEYS      0u                       // u64[131072]  -> 1,048,576
#define WS_PROPS_ALL 1048576u                 // float4[90000]-> +1,440,000
#define WS_PROPS_TOP 2488576u                 // float4[6000] -> +96,000
#define WS_AREAS_TOP 2584576u                 // float[6000]  -> +24,000
#define WS_MASK      2608576u                 // u32[6000*188]-> +4,512,000
#define WS_SEL       7120576u                 // int[300]
#define WS_COUNT     7121792u                 // int

// ---------------- optional gfx1250 async-to-LDS path ----------------------
#if defined(__has_builtin)
#if __has_builtin(__builtin_amdgcn_global_load_async_to_lds_b64)
#define HAVE_ASYNC_LDS 1
#endif
#if __has_builtin(__builtin_amdgcn_s_wait_asynccnt)
#define HAVE_WAIT_ASYNC_BUILTIN 1
#endif
#endif

__device__ __forceinline__ void wait_asynccnt0() {
#if defined(HAVE_WAIT_ASYNC_BUILTIN)
  __builtin_amdgcn_s_wait_asynccnt(0);
#else
  asm volatile("s_wait_asynccnt 0" ::: "memory");
#endif
}

// ---------------- kernel 1: anchors + proposals + sort keys ----------------
__global__ __launch_bounds__(256)
void proposals_kernel(const float* __restrict__ scores,
                      const float* __restrict__ deltas,
                      const float* __restrict__ im_info,
                      AnchorsArg anch,
                      float* __restrict__ out_anchors,      // d_out + 1500
                      float4* __restrict__ props_all,
                      unsigned long long* __restrict__ keys) {
  int idx = blockIdx.x * blockDim.x + threadIdx.x;
  if (idx >= NPAD) return;
  if (idx >= NPROP) { keys[idx] = 0ull; return; }

  int a = idx % NA;
  int s = idx / NA;
  int w = s % W_FEAT;
  int h = s / W_FEAT;

  float shx = (float)(w * 16);
  float shy = (float)(h * 16);
  float ax1 = anch.v[a * 4 + 0] + shx;
  float ay1 = anch.v[a * 4 + 1] + shy;
  float ax2 = anch.v[a * 4 + 2] + shx;
  float ay2 = anch.v[a * 4 + 3] + shy;

  out_anchors[idx * 4 + 0] = ax1;
  out_anchors[idx * 4 + 1] = ay1;
  out_anchors[idx * 4 + 2] = ax2;
  out_anchors[idx * 4 + 3] = ay2;

  float aw = ax2 - ax1 + 1.0f;
  float ah = ay2 - ay1 + 1.0f;
  float cx = ax1 + 0.5f * aw;
  float cy = ay1 + 0.5f * ah;

  int base = h * W_FEAT + w;
  float dx = deltas[(4 * a + 0) * HWSZ + base];
  float dy = deltas[(4 * a + 1) * HWSZ + base];
  float dw = deltas[(4 * a + 2) * HWSZ + base];
  float dh = deltas[(4 * a + 3) * HWSZ + base];

  float pcx = dx * aw + cx;
  float pcy = dy * ah + cy;
  float pw  = expf(dw) * aw;
  float ph  = expf(dh) * ah;

  float x1 = pcx - 0.5f * pw;
  float y1 = pcy - 0.5f * ph;
  float x2 = pcx + 0.5f * pw;
  float y2 = pcy + 0.5f * ph;

  float mx = im_info[1] - 1.0f;
  float my = im_info[0] - 1.0f;
  x1 = fminf(fmaxf(x1, 0.0f), mx);
  y1 = fminf(fmaxf(y1, 0.0f), my);
  x2 = fminf(fmaxf(x2, 0.0f), mx);
  y2 = fminf(fmaxf(y2, 0.0f), my);

  props_all[idx] = make_float4(x1, y1, x2, y2);

  float sc = scores[(NA + a) * HWSZ + base];
  unsigned sb = __float_as_uint(sc);   // scores are non-negative: bits monotonic
  keys[idx] = ((unsigned long long)sb << 32) |
              (unsigned long long)(0xFFFFFFFFu - (unsigned)idx);
}

// ---------------- kernel 2a: global bitonic step (span >= 2048) -----------
__global__ __launch_bounds__(256)
void bitonic_step(unsigned long long* __restrict__ keys, int k, int j) {
  int i = blockIdx.x * blockDim.x + threadIdx.x;
  int l = i ^ j;
  if (l > i) {
    unsigned long long a = keys[i];
    unsigned long long b = keys[l];
    bool dirDesc = ((i & k) == 0);
    if ((a < b) == dirDesc) { keys[i] = b; keys[l] = a; }
  }
}

// ---------------- kernel 2b: fused LDS bitonic steps (span <= 1024) -------
// One block sorts a 2048-key tile through all remaining j-steps of stage k.
// Tile load uses the gfx1250 async DMA path (global_load_async_to_lds_b64).
__global__ __launch_bounds__(256)
void bitonic_fused(unsigned long long* __restrict__ keys, int k, int jstart) {
  __shared__ unsigned long long sh[2048];
  int base = blockIdx.x * 2048;

#if defined(HAVE_ASYNC_LDS)
  for (int t = threadIdx.x; t < 2048; t += 256) {
    gptr_v2i gsrc = (gptr_v2i)(uintptr_t)(keys + base + t);
    lptr_v2i ldst = (lptr_v2i)(unsigned)(uintptr_t)(&sh[t]);
    __builtin_amdgcn_global_load_async_to_lds_b64(gsrc, ldst, 0, 0);
  }
  wait_asynccnt0();
  __syncthreads();
#else
  for (int t = threadIdx.x; t < 2048; t += 256) sh[t] = keys[base + t];
  __syncthreads();
#endif

  for (int j = jstart; j >= 1; j >>= 1) {
    for (int c = threadIdx.x; c < 1024; c += 256) {
      int i = ((c / j) * (j << 1)) + (c % j);
      int l = i + j;
      bool dirDesc = (((base + i) & k) == 0);
      unsigned long long a = sh[i];
      unsigned long long b = sh[l];
      if ((a < b) == dirDesc) { sh[i] = b; sh[l] = a; }
    }
    __syncthreads();
  }

  for (int t = threadIdx.x; t < 2048; t += 256) keys[base + t] = sh[t];
}

// ---------------- kernel 3: gather top-6000 ----------------
__global__ __launch_bounds__(256)
void topk_gather_kernel(const unsigned long long* __restrict__ keys,
                        const float4* __restrict__ props_all,
                        float4* __restrict__ props_top,
                        float*  __restrict__ areas_top) {
  int t = blockIdx.x * blockDim.x + threadIdx.x;
  if (t >= NSEL) return;
  unsigned long long key = keys[t];
  unsigned idx = 0xFFFFFFFFu - (unsigned)(key & 0xFFFFFFFFull);
  float4 p = props_all[idx];
  props_top[t] = p;
  areas_top[t] = (p.z - p.x + 1.0f) * (p.w - p.y + 1.0f);
}

// ---------------- kernel 4: NMS suppression bitmask (wave32 + WMMA) -------
// One wave computes a 16-row x 32-col tile of the IoU>thresh matrix.
// area_i + area_j (the IoU denominator's additive part) is computed as a
// rank-2 outer sum via V_WMMA_F32_16X16X4_F32: A=[area_i,1,0,0], B=[1;area_j;0;0].
__global__ __launch_bounds__(256)
void nms_mask_kernel(const float4* __restrict__ props,
                     const float*  __restrict__ areas,
                     unsigned*     __restrict__ mask) {
  int wave = blockIdx.x * (blockDim.x >> 5) + (threadIdx.x >> 5);
  int lane = threadIdx.x & 31;
  if (wave >= RB * NW) return;

  int r  = wave / NW;        // row tile (16 rows)
  int wc = wave % NW;        // 32-bit word column
  int i0 = r * 16;
  int j0 = wc * 32;
  int n  = lane & 15;
  bool hi = lane >= 16;

  __builtin_prefetch(props + i0, 0, 0);   // global_prefetch_b8

  int jA = j0 + n;        int jAc = jA < NSEL ? jA : (NSEL - 1);
  int jB = j0 + 16 + n;   int jBc = jB < NSEL ? jB : (NSEL - 1);
  float4 cbA = props[jAc];
  float4 cbB = props[jBc];
  float  caA = areas[jAc];
  float  caB = areas[jBc];
  float  arow = areas[i0 + n];   // valid: i0+15 <= 5999; unused on hi lanes

  // A 16x4 f32: lanes 0-15 hold K0,K1 for row M=lane; lanes 16-31 hold K2,K3 (=0)
  v2f A;  A.x  = hi ? 0.0f : arow;  A.y  = hi ? 0.0f : 1.0f;
  // B 4x16 f32: lanes 0-15 hold K0,K1 for col N=lane; lanes 16-31 hold K2,K3 (=0)
  v2f BA; BA.x = hi ? 0.0f : 1.0f;  BA.y = hi ? 0.0f : caA;
  v2f BB; BB.x = hi ? 0.0f : 1.0f;  BB.y = hi ? 0.0f : caB;
  v8f z = {0.f, 0.f, 0.f, 0.f, 0.f, 0.f, 0.f, 0.f};

  // D[m][n] = area_row[m] + area_col[n]
  v8f dA = __builtin_amdgcn_wmma_f32_16x16x4_f32(false, A, false, BA, (short)0, z, false, false);
  v8f dB = __builtin_amdgcn_wmma_f32_16x16x4_f32(false, A, false, BB, (short)0, z, false, false);

#pragma unroll
  for (int k = 0; k < 8; ++k) {
    int m = k + (hi ? 8 : 0);          // matches C/D VGPR-k row mapping
    int i = i0 + m;
    float4 rb = props[i];

    float iwA = fmaxf(fminf(rb.z, cbA.z) - fmaxf(rb.x, cbA.x) + 1.0f, 0.0f);
    float ihA = fmaxf(fminf(rb.w, cbA.w) - fmaxf(rb.y, cbA.y) + 1.0f, 0.0f);
    float interA = iwA * ihA;
    float iouA = interA / (dA[k] - interA);
    bool pA = (iouA > NMS_TH) && (jA > i) && (jA < NSEL);
    unsigned bA = __builtin_amdgcn_ballot_w32(pA);

    float iwB = fmaxf(fminf(rb.z, cbB.z) - fmaxf(rb.x, cbB.x) + 1.0f, 0.0f);
    float ihB = fmaxf(fminf(rb.w, cbB.w) - fmaxf(rb.y, cbB.y) + 1.0f, 0.0f);
    float interB = iwB * ihB;
    float iouB = interB / (dB[k] - interB);
    bool pB = (iouB > NMS_TH) && (jB > i) && (jB < NSEL);
    unsigned bB = __builtin_amdgcn_ballot_w32(pB);

    if (lane == 0) {
      // bA bits[15:0]=row k cols j0..j0+15, bits[31:16]=row k+8 same cols
      mask[(i0 + k)     * NW + wc] = (bA & 0xFFFFu) | (bB << 16);
      mask[(i0 + k + 8) * NW + wc] = (bA >> 16)     | (bB & 0xFFFF0000u);
    }
  }
}

// ---------------- kernel 5: greedy reduction, single wave32, no barriers --
// Lane L owns suppression words {q*32+L : q<6} in registers. The keep bit is
// broadcast via __shfl (ds_bpermute); control flow stays wave-uniform.
__global__ __launch_bounds__(32)
void nms_reduce_kernel(const unsigned* __restrict__ mask,
                       int* __restrict__ sel, int* __restrict__ count) {
  int lane = threadIdx.x;
  unsigned rem[6] = {0u, 0u, 0u, 0u, 0u, 0u};
  int cnt = 0;

#pragma unroll
  for (int slot = 0; slot < 6; ++slot) {     // words slot*32 .. slot*32+31
    int i_end = (slot + 1) * 1024;
    if (i_end > NSEL) i_end = NSEL;
    for (int i = slot * 1024; i < i_end; ++i) {
      unsigned wv = __shfl(rem[slot], (i >> 5) & 31, 32);  // uniform broadcast
      bool keep = ((wv >> (i & 31)) & 1u) == 0u;
      if (keep) {                                          // wave-uniform
#pragma unroll
        for (int q = 0; q < 6; ++q) {
          int w = q * 32 + lane;
          if (w < NW) rem[q] |= mask[i * NW + w];
        }
        if (lane == 0) { if (cnt < NPOST) sel[cnt] = i; cnt++; }
      }
    }
  }
  if (lane == 0) *count = cnt > NPOST ? NPOST : cnt;
}

// ---------------- kernel 6: write output rows -----------------------------
__global__ __launch_bounds__(256)
void output_kernel(const float4* __restrict__ props,
                   const int* __restrict__ sel,
                   const int* __restrict__ count,
                   float* __restrict__ out) {
  int r = blockIdx.x * blockDim.x + threadIdx.x;
  if (r >= NPOST) return;
  float4 b = make_float4(0.f, 0.f, 0.f, 0.f);
  int c = *count;
  if (r < c) b = props[sel[r]];
  out[r * 5 + 0] = 0.0f;
  out[r * 5 + 1] = b.x;
  out[r * 5 + 2] = b.y;
  out[r * 5 + 3] = b.z;
  out[r * 5 + 4] = b.w;
}

// ---------------- host-side base anchors (numpy semantics) ----------------
static void make_anchors(AnchorsArg* A) {
  const double ratios[3] = {0.5, 1.0, 2.0};
  const double scales[3] = {8.0, 16.0, 32.0};
  const double base_size = 16.0;
  double w = base_size, h = base_size, xc = 0.5 * (w - 1.0), yc = 0.5 * (h - 1.0);
  double size = w * h;
  int k = 0;
  for (int ri = 0; ri < 3; ++ri) {
    double ws = nearbyint(sqrt(size / ratios[ri]));   // ties-to-even == np.round
    double hs = nearbyint(ws * ratios[ri]);
    double ax1 = xc - 0.5 * (ws - 1.0), ay1 = yc - 0.5 * (hs - 1.0);
    double ax2 = xc + 0.5 * (ws - 1.0), ay2 = yc + 0.5 * (hs - 1.0);
    double aw = ax2 - ax1 + 1.0, ah = ay2 - ay1 + 1.0;
    double axc = ax1 + 0.5 * (aw - 1.0), ayc = ay1 + 0.5 * (ah - 1.0);
    for (int si = 0; si < 3; ++si) {
      double sw = aw * scales[si], sh = ah * scales[si];
      A->v[k * 4 + 0] = (float)(axc - 0.5 * (sw - 1.0));
      A->v[k * 4 + 1] = (float)(ayc - 0.5 * (sh - 1.0));
      A->v[k * 4 + 2] = (float)(axc + 0.5 * (sw - 1.0));
      A->v[k * 4 + 3] = (float)(ayc + 0.5 * (sh - 1.0));
      ++k;
    }
  }
}

// ---------------- launcher ------------------------------------------------
extern "C" void kernel_launch(void* const* d_in, const int* in_sizes, int n_in,
                              void* d_out, int out_size, void* d_ws, size_t ws_size,
                              hipStream_t stream) {
  const float* scores  = (const float*)d_in[0];
  const float* deltas  = (const float*)d_in[1];
  const float* im_info = (const float*)d_in[2];
  (void)in_sizes; (void)n_in; (void)out_size; (void)ws_size;

  float* out = (float*)d_out;                 // [300*5 output][90000*4 anchors]
  char* ws = (char*)d_ws;
  unsigned long long* keys = (unsigned long long*)(ws + WS_KEYS);
  float4* props_all = (float4*)(ws + WS_PROPS_ALL);
  float4* props_top = (float4*)(ws + WS_PROPS_TOP);
  float*  areas_top = (float*) (ws + WS_AREAS_TOP);
  unsigned* mask    = (unsigned*)(ws + WS_MASK);
  int* sel          = (int*)(ws + WS_SEL);
  int* count        = (int*)(ws + WS_COUNT);

  AnchorsArg anch;
  make_anchors(&anch);

  // 1) proposals + anchors + keys (incl. pad)
  proposals_kernel<<<NPAD / 256, 256, 0, stream>>>(
      scores, deltas, im_info, anch, out + NPOST * 5, props_all, keys);

  // 2) bitonic sort, descending, 131072 u64 keys
  //    far steps (j >= 2048) in global memory, rest fused in 16KB LDS tiles
  for (int k = 2; k <= NPAD; k <<= 1) {
    int j = k >> 1;
    for (; j >= 2048; j >>= 1)
      bitonic_step<<<NPAD / 256, 256, 0, stream>>>(keys, k, j);
    bitonic_fused<<<NPAD / 2048, 256, 0, stream>>>(keys, k, j);  // j <= 1024
  }

  // 3) gather top 6000
  topk_gather_kernel<<<(NSEL + 255) / 256, 256, 0, stream>>>(
      keys, props_all, props_top, areas_top);

  // 4) pairwise suppression bitmask (one wave per 16x32 tile)
  {
    int tiles = RB * NW;                   // 70500
    int waves_per_block = 256 / 32;        // 8
    int blocks = (tiles + waves_per_block - 1) / waves_per_block;
    nms_mask_kernel<<<blocks, 256, 0, stream>>>(props_top, areas_top, mask);
  }

  // 5) greedy scan (single wave32, barrier-free)
  nms_reduce_kernel<<<1, 32, 0, stream>>>(mask, sel, count);

  // 6) emit the 300x5 output block
  output_kernel<<<(NPOST + 255) / 256, 256, 0, stream>>>(props_top, sel, count, out);
}